// LNCC_3934190044170
// MI455X (gfx1250) — compile-verified
//
#include <hip/hip_runtime.h>
#include <hip/hip_bf16.h>

// LNCC on MI455X (gfx1250, wave32).
// Pass 1: z-direction 5-tap sliding sums of {M, R, M^2, R^2, M*R} -> 5 ws fields.
// Pass 2: per (z, x-strip) workgroup (1 wave): x-sum and y-sum done as banded
//         16x16xK f32 WMMA matmuls (band weights are exact 0/1), then the
//         LNCC pointwise formula and a wave reduction -> per-block partial.
// Pass 3: deterministic tree reduction of partials -> -mean(corr).

#define DXYZ 160
#define PLANE (DXYZ * DXYZ)          // 25600
#define VOL   (DXYZ * DXYZ * DXYZ)   // 4096000
#define NB    2
#define NVOX  (NB * VOL)             // 8192000
#define NTILE (DXYZ / 16)            // 10
#define NBLK2 (NTILE * DXYZ * NB)    // 3200 pass-2 workgroups

typedef __attribute__((ext_vector_type(2))) float v2f;
typedef __attribute__((ext_vector_type(8))) float v8f;

// ---------------------------------------------------------------------------
// Pass 1: sliding-window sums along z. One thread per (b,y,x) column.
// Window radius 2, zero padding outside [0,160) (matches conv zero-pad).
// ---------------------------------------------------------------------------
__global__ __launch_bounds__(DXYZ)
void lncc_zsum(const float* __restrict__ M, const float* __restrict__ R,
               float* __restrict__ wsM, float* __restrict__ wsR,
               float* __restrict__ wsMM, float* __restrict__ wsRR,
               float* __restrict__ wsMR) {
    const int x = threadIdx.x;        // 0..159
    const int y = blockIdx.x;         // 0..159
    const int b = blockIdx.y;         // 0..1
    const size_t base = (size_t)b * VOL + (size_t)y * DXYZ + x;  // z = 0

    float mh[5], rh[5];               // window values for z-2 .. z+2
    mh[0] = mh[1] = 0.f; rh[0] = rh[1] = 0.f;
    mh[2] = M[base];                 rh[2] = R[base];
    mh[3] = M[base + PLANE];         rh[3] = R[base + PLANE];
    mh[4] = M[base + 2 * (size_t)PLANE]; rh[4] = R[base + 2 * (size_t)PLANE];

    for (int z = 0; z < DXYZ; ++z) {
        float sm = 0.f, sr = 0.f, smm = 0.f, srr = 0.f, smr = 0.f;
#pragma unroll
        for (int i = 0; i < 5; ++i) {
            const float m = mh[i], r = rh[i];
            sm += m; sr += r; smm += m * m; srr += r * r; smr += m * r;
        }
        const size_t idx = base + (size_t)z * PLANE;
        wsM[idx] = sm; wsR[idx] = sr; wsMM[idx] = smm; wsRR[idx] = srr; wsMR[idx] = smr;
#pragma unroll
        for (int i = 0; i < 4; ++i) { mh[i] = mh[i + 1]; rh[i] = rh[i + 1]; }
        const int zn = z + 3;
        if (zn < DXYZ) {
            mh[4] = M[base + (size_t)zn * PLANE];
            rh[4] = R[base + (size_t)zn * PLANE];
        } else {
            mh[4] = 0.f; rh[4] = 0.f;
        }
    }
}

// ---------------------------------------------------------------------------
// Pass 2: one wave (32 threads) per (tx, z, b). x-sum and y-sum via banded
// f32 WMMA; LNCC formula; wave-reduced partial written per block.
//
// WMMA f32 16x16x4 operand layouts (ISA 7.12.2, wave32):
//   A (16x4,  MxK): lanes 0-15: M=lane,    v0->K=k0+0, v1->K=k0+1
//                   lanes16-31: M=lane-16, v0->K=k0+2, v1->K=k0+3
//   B (4x16,  KxN): mirrors A with N=lane%16 in place of M.
//   C/D (16x16):    VGPR v: lanes 0-15 -> (M=v,   N=lane)
//                           lanes16-31 -> (M=v+8, N=lane-16)
// Band matrix for a radius-2 box sum with 2-left-halo indexing:
//   w[k][n] = 1  iff  n <= k <= n+4   (values computed per-lane, no memory)
// ---------------------------------------------------------------------------
__global__ __launch_bounds__(32)
void lncc_xysum(const float* __restrict__ wsM, const float* __restrict__ wsR,
                const float* __restrict__ wsMM, const float* __restrict__ wsRR,
                const float* __restrict__ wsMR, const float* __restrict__ kern,
                float* __restrict__ partial) {
    __shared__ float in_s[DXYZ][20];        // one field, rows y, cols x-halo (12.8 KB)
    __shared__ float xs_s[5][DXYZ][16];     // x-summed fields               (51.2 KB)

    const int tx = blockIdx.x;              // 0..9   (16-wide x strip)
    const int z  = blockIdx.y;              // 0..159
    const int b  = blockIdx.z;              // 0..1
    const int l  = threadIdx.x;             // 0..31
    const bool hi = (l >= 16);
    const int lo = l & 15;

    const float* fields[5] = { wsM, wsR, wsMM, wsRR, wsMR };
    const size_t planeBase = ((size_t)b * DXYZ + z) * PLANE;

#pragma unroll
    for (int f = 0; f < 5; ++f) {
        const float* __restrict__ src = fields[f];
        // Stage 160 x 20 halo region (zero-padded in x) into LDS.
        for (int i = l; i < DXYZ * 20; i += 32) {
            const int y = i / 20, c = i % 20;
            const int x = 16 * tx - 2 + c;
            in_s[y][c] = (x >= 0 && x < DXYZ)
                           ? src[planeBase + (size_t)y * DXYZ + x] : 0.f;
        }
        __syncthreads();

        // x-direction 5-tap sum: D(16x16) = A_data(16x20) x B_band(20x16)
        for (int ty = 0; ty < NTILE; ++ty) {
            v8f C = {};
#pragma unroll
            for (int c5 = 0; c5 < 5; ++c5) {        // K = 20 in chunks of 4
                const int ka = c5 * 4 + (hi ? 2 : 0);
                v2f a, bb;
                a.x = in_s[16 * ty + lo][ka];
                a.y = in_s[16 * ty + lo][ka + 1];
                bb.x = (ka     >= lo && ka     <= lo + 4) ? 1.f : 0.f;
                bb.y = (ka + 1 >= lo && ka + 1 <= lo + 4) ? 1.f : 0.f;
                C = __builtin_amdgcn_wmma_f32_16x16x4_f32(
                        false, a, false, bb, (short)0, C, false, false);
            }
#pragma unroll
            for (int v = 0; v < 8; ++v)
                xs_s[f][16 * ty + v + (hi ? 8 : 0)][lo] = C[v];
        }
        __syncthreads();   // done with in_s before next field reloads it
    }

    // y-direction 5-tap sum + LNCC formula.
    const float k0 = kern[0];               // 1/125 (uniform kernel weight)
    float acc = 0.f;
    for (int ty = 0; ty < NTILE; ++ty) {
        v8f E[5];
#pragma unroll
        for (int f = 0; f < 5; ++f) {
            v8f C = {};
#pragma unroll
            for (int c5 = 0; c5 < 5; ++c5) {    // K = 20: rows 16*ty-2 .. 16*ty+17
                const int ka = c5 * 4 + (hi ? 2 : 0);
                v2f a, bb;
                a.x = (ka     >= lo && ka     <= lo + 4) ? 1.f : 0.f;
                a.y = (ka + 1 >= lo && ka + 1 <= lo + 4) ? 1.f : 0.f;
                const int y0 = 16 * ty - 2 + ka;
                const int y1 = y0 + 1;
                bb.x = (y0 >= 0 && y0 < DXYZ) ? xs_s[f][y0][lo] : 0.f;
                bb.y = (y1 >= 0 && y1 < DXYZ) ? xs_s[f][y1][lo] : 0.f;
                C = __builtin_amdgcn_wmma_f32_16x16x4_f32(
                        false, a, false, bb, (short)0, C, false, false);
            }
            E[f] = C;
        }
#pragma unroll
        for (int v = 0; v < 8; ++v) {
            const float sm  = E[0][v] * k0;
            const float sr  = E[1][v] * k0;
            const float smm = E[2][v] * k0;
            const float srr = E[3][v] * k0;
            const float smr = E[4][v] * k0;
            const float mv = sqrtf(smm - sm * sm + 1e-5f);
            const float rv = sqrtf(srr - sr * sr + 1e-5f);
            acc += (smr - sm * sr) / (mv * rv + 1e-5f);
        }
    }

    // Wave32 reduction, then one partial per block (deterministic final pass).
#pragma unroll
    for (int off = 16; off > 0; off >>= 1)
        acc += __shfl_down(acc, off, 32);
    if (l == 0) {
        const int bid = (blockIdx.z * gridDim.y + blockIdx.y) * gridDim.x + blockIdx.x;
        partial[bid] = acc;
    }
}

// ---------------------------------------------------------------------------
// Pass 3: deterministic tree reduction of 3200 partials -> -mean(corr).
// ---------------------------------------------------------------------------
__global__ __launch_bounds__(256)
void lncc_finish(const float* __restrict__ partial, float* __restrict__ out, int n) {
    __shared__ float red[256];
    float s = 0.f;
    for (int i = threadIdx.x; i < n; i += 256) s += partial[i];
    red[threadIdx.x] = s;
    __syncthreads();
#pragma unroll
    for (int st = 128; st > 0; st >>= 1) {
        if (threadIdx.x < st) red[threadIdx.x] += red[threadIdx.x + st];
        __syncthreads();
    }
    if (threadIdx.x == 0) out[0] = -red[0] * (1.f / (float)NVOX);
}

extern "C" void kernel_launch(void* const* d_in, const int* in_sizes, int n_in,
                              void* d_out, int out_size, void* d_ws, size_t ws_size,
                              hipStream_t stream) {
    const float* M    = (const float*)d_in[0];
    const float* R    = (const float*)d_in[1];
    const float* kern = (const float*)d_in[2];
    float* out = (float*)d_out;

    // Workspace layout: 5 fields of NVOX floats (163.84 MB) + 3200 partials.
    float* ws   = (float*)d_ws;
    float* wsM  = ws + 0UL * NVOX;
    float* wsR  = ws + 1UL * NVOX;
    float* wsMM = ws + 2UL * NVOX;
    float* wsRR = ws + 3UL * NVOX;
    float* wsMR = ws + 4UL * NVOX;
    float* part = ws + 5UL * NVOX;

    lncc_zsum<<<dim3(DXYZ, NB), dim3(DXYZ), 0, stream>>>(
        M, R, wsM, wsR, wsMM, wsRR, wsMR);

    lncc_xysum<<<dim3(NTILE, DXYZ, NB), dim3(32), 0, stream>>>(
        wsM, wsR, wsMM, wsRR, wsMR, kern, part);

    lncc_finish<<<dim3(1), dim3(256), 0, stream>>>(part, out, NBLK2);
}